// CNF1D_88845693485257
// MI455X (gfx1250) — compile-verified
//
#include <hip/hip_runtime.h>
#include <math.h>

typedef __attribute__((ext_vector_type(2))) float v2f;
typedef __attribute__((ext_vector_type(8))) float v8f;

#define HDIM     256
#define KCH      64          // 256 / 4 K-chunks for 16x16x4 WMMA
#define TS       16          // samples per workgroup (one M-tile)
#define NTHREADS 512         // 16 waves: wave w owns N-tile w

// LDS layout (float offsets)
#define OFF_W2   0
#define OFF_H1   (HDIM*HDIM)               // 65536
#define OFF_H1D  (OFF_H1 + TS*HDIM)        // +4096
#define OFF_W1Z  (OFF_H1D + TS*HDIM)       // +4096
#define OFF_W1T  (OFF_W1Z + HDIM)
#define OFF_B1   (OFF_W1T + HDIM)
#define OFF_B2   (OFF_B1 + HDIM)
#define OFF_W3   (OFF_B2 + HDIM)
#define OFF_F    (OFF_W3 + HDIM)
#define OFF_FD   (OFF_F + TS)
#define OFF_Z    (OFF_FD + TS)
#define OFF_ZT   (OFF_Z + TS)
#define OFF_DIV  (OFF_ZT + TS)
#define OFF_KZ   (OFF_DIV + TS)
#define OFF_KD   (OFF_KZ + TS)
#define LDS_FLOATS (OFF_KD + TS)           // 75120 floats = 300480 B <= 320 KB

__device__ __forceinline__ float tanh_fast(float x) {
#if defined(__AMDGCN__) && __has_builtin(__builtin_amdgcn_tanhf)
  return __builtin_amdgcn_tanhf(x);        // gfx1250 v_tanh_f32
#else
  float e = __expf(2.0f * x);
  return 1.0f - 2.0f / (e + 1.0f);
#endif
}

__global__ void __launch_bounds__(NTHREADS, 1)
cnf1d_rk4_wmma(const float* __restrict__ z0, const float* __restrict__ W1,
               const float* __restrict__ b1, const float* __restrict__ W2,
               const float* __restrict__ b2, const float* __restrict__ W3,
               const float* __restrict__ b3, float* __restrict__ out, int Btot)
{
  extern __shared__ float sm[];
  float* sW2  = sm + OFF_W2;
  float* sH1  = sm + OFF_H1;
  float* sH1d = sm + OFF_H1D;
  float* sW1z = sm + OFF_W1Z;
  float* sW1t = sm + OFF_W1T;
  float* sB1  = sm + OFF_B1;
  float* sB2  = sm + OFF_B2;
  float* sW3  = sm + OFF_W3;
  float* sF   = sm + OFF_F;
  float* sFd  = sm + OFF_FD;
  float* sZ   = sm + OFF_Z;
  float* sZt  = sm + OFF_ZT;
  float* sDiv = sm + OFF_DIV;
  float* sKz  = sm + OFF_KZ;
  float* sKd  = sm + OFF_KD;

  const int tid  = threadIdx.x;
  const int lane = tid & 31;
  const int wv   = tid >> 5;        // this wave's N-tile (0..15)
  const float dtv = 0.25f;          // (T1-T0)/N_STEPS
  const float b3v = b3[0];

  // ---- one-time loads into LDS ----
  for (int i = tid; i < HDIM; i += NTHREADS) {
    sW1z[i] = W1[i];                // W1[0,:] (z column)
    sW1t[i] = W1[HDIM + i];         // W1[1,:] (t column)
    sB1[i]  = b1[i];
    sB2[i]  = b2[i];
    sW3[i]  = W3[i];
  }
  // W2 (row-major 256x256) -> B-fragment layout for V_WMMA_F32_16X16X4_F32:
  // frag(c=kchunk,n=ntile): lane l, dword d holds W2[4c + 2*(l>>4) + d][16n + (l&15)]
  for (int e = tid; e < HDIM * HDIM; e += NTHREADS) {
    int row = e >> 8, col = e & 255;
    int c  = row >> 2, rr = row & 3;
    int n  = col >> 4, lo = col & 15;
    int l  = ((rr >> 1) << 4) | lo;
    int d  = rr & 1;
    int idx = ((((((c << 4) | n) << 5) | l) << 1) | d);
    sW2[idx] = W2[e];
  }
  if (tid < TS) {
    float z = z0[(size_t)blockIdx.x * TS + tid];
    sZ[tid] = z; sZt[tid] = z; sDiv[tid] = 0.0f;
    sKz[tid] = 0.0f; sKd[tid] = 0.0f;
    sF[tid] = b3v; sFd[tid] = 0.0f;
  }
  __syncthreads();

  // ---- 4 RK4 steps x 4 stages = 16 vector-field (+JVP) evaluations ----
  #pragma unroll 1
  for (int ev = 0; ev < 16; ++ev) {
    const int stg = ev & 3;
    float tt = (float)(ev >> 2) * dtv;
    if (stg == 3) tt += dtv; else if (stg != 0) tt += 0.5f * dtv;

    // stage 1: h1 = tanh(z*W1z + t*W1t + b1), h1d = (1-h1^2)*W1z
    // written directly in A-fragment layout: slot s = ((c*32 + l)*2 + d)
    #pragma unroll
    for (int i = 0; i < 8; ++i) {
      int s = tid + NTHREADS * i;          // 0..4095, lane-contiguous stores
      int d = s & 1;
      int l = (s >> 1) & 31;
      int c = s >> 6;
      int m = l & 15;
      int j = (c << 2) + ((l >> 4) << 1) + d;
      float w1 = sW1z[j];
      float a  = sZt[m] * w1 + tt * sW1t[j] + sB1[j];
      float h  = tanh_fast(a);
      sH1[s]  = h;
      sH1d[s] = (1.0f - h * h) * w1;
    }
    __syncthreads();

    // stage 2: a2 = h1 @ W2, a2d = h1d @ W2 for this wave's 16 columns
    v8f acc  = {};
    v8f accd = {};
    const v2f* aF  = (const v2f*)sH1;
    const v2f* adF = (const v2f*)sH1d;
    const v2f* bF  = (const v2f*)sW2;
    #pragma unroll 4
    for (int c = 0; c < KCH; ++c) {
      v2f af = aF[(c << 5) + lane];
      v2f ad = adF[(c << 5) + lane];
      v2f bf = bF[((((c << 4) | wv) << 5) + lane)];
      acc  = __builtin_amdgcn_wmma_f32_16x16x4_f32(false, af, false, bf,
                                                   (short)0, acc,  false, false);
      accd = __builtin_amdgcn_wmma_f32_16x16x4_f32(false, ad, false, bf,
                                                   (short)0, accd, false, false);
    }

    // stage 3: h2 = tanh(a2+b2); f = h2@W3 (+b3 preloaded), fd = (1-h2^2)*a2d@W3
    {
      const int n  = lane & 15;
      const int j  = (wv << 4) | n;
      const int mb = (lane >> 4) << 3;     // C layout: m = v + 8*(lane>=16)
      const float w3 = sW3[j];
      const float bb = sB2[j];
      #pragma unroll
      for (int v = 0; v < 8; ++v) {
        float h2  = tanh_fast(acc[v] + bb);
        float pf  = h2 * w3;
        float pfd = (1.0f - h2 * h2) * accd[v] * w3;
        #pragma unroll
        for (int msk = 1; msk < 16; msk <<= 1) {   // reduce over 16 N-lanes
          pf  += __shfl_xor(pf,  msk, 32);
          pfd += __shfl_xor(pfd, msk, 32);
        }
        if (n == 0) {
          atomicAdd(&sF[mb + v],  pf);     // ds_add_f32, 16 adds per address
          atomicAdd(&sFd[mb + v], pfd);
        }
      }
    }
    __syncthreads();

    // stage 4: RK4 bookkeeping (16 scalar lanes)
    if (tid < TS) {
      float kz = sF[tid];
      float kd = sFd[tid];
      float w  = (stg == 1 || stg == 2) ? 2.0f : 1.0f;
      float akz = sKz[tid] + w * kz;
      float akd = sKd[tid] + w * kd;
      if (stg < 3) {
        float cf = (stg == 2) ? dtv : 0.5f * dtv;
        sZt[tid] = sZ[tid] + cf * kz;
        sKz[tid] = akz; sKd[tid] = akd;
      } else {
        float zn = sZ[tid] + (dtv * (1.0f / 6.0f)) * akz;
        sZ[tid] = zn; sZt[tid] = zn;
        sDiv[tid] += (dtv * (1.0f / 6.0f)) * akd;
        sKz[tid] = 0.0f; sKd[tid] = 0.0f;
      }
      sF[tid] = b3v; sFd[tid] = 0.0f;      // reset accumulators for next eval
    }
    __syncthreads();
  }

  if (tid < TS) {
    int g = blockIdx.x * TS + tid;
    out[g] = sZ[tid];                      // zf
    out[Btot + g] = sDiv[tid];             // integrated divergence
  }
}

extern "C" void kernel_launch(void* const* d_in, const int* in_sizes, int n_in,
                              void* d_out, int out_size, void* d_ws, size_t ws_size,
                              hipStream_t stream) {
  (void)n_in; (void)d_ws; (void)ws_size; (void)out_size;
  const float* z0 = (const float*)d_in[0];
  const float* W1 = (const float*)d_in[1];
  const float* b1 = (const float*)d_in[2];
  const float* W2 = (const float*)d_in[3];
  const float* b2 = (const float*)d_in[4];
  const float* W3 = (const float*)d_in[5];
  const float* b3 = (const float*)d_in[6];
  float* out = (float*)d_out;

  const int B    = in_sizes[0];            // 32768
  const int nblk = B / TS;                 // 2048 workgroups
  const size_t lds = (size_t)LDS_FLOATS * sizeof(float);  // ~300 KB (CDNA5: 320 KB/WGP)

  hipFuncSetAttribute(reinterpret_cast<const void*>(cnf1d_rk4_wmma),
                      hipFuncAttributeMaxDynamicSharedMemorySize, (int)lds);
  cnf1d_rk4_wmma<<<nblk, NTHREADS, lds, stream>>>(z0, W1, b1, W2, b2, W3, b3, out, B);
}